// VDBGrid_54073638256774
// MI455X (gfx1250) — compile-verified
//
#include <hip/hip_runtime.h>

// Trilinear interpolation of a 160x160x160x12 f32 grid at 2M points.
// Memory-bound gather: optimize for L2 residency of the grid (196.6 MB vs
// 192 MB L2) by streaming xyz/out with non-temporal hints, and for memory-
// level parallelism (24 outstanding global_load_b128 per lane).

typedef float v4f __attribute__((ext_vector_type(4)));

namespace {
constexpr int WX = 160, WY = 160, WZ = 160, CH = 12;
constexpr int SX = WY * WZ * CH;   // x stride in floats = 307200
constexpr int SY = WZ * CH;        // y stride in floats = 1920
constexpr int SZ = CH;             // z stride in floats = 12
}

__global__ __launch_bounds__(256) void vdb_trilerp12(
    const float* __restrict__ xyz,
    const float* __restrict__ grid,
    float* __restrict__ out,
    int n)
{
    const int i = blockIdx.x * blockDim.x + threadIdx.x;
    if (i >= n) return;

    // xyz is read exactly once -> non-temporal loads (don't pollute L2;
    // keep the grid resident there).
    const size_t b3 = 3ull * (unsigned)i;
    float px = __builtin_nontemporal_load(xyz + b3 + 0);
    float py = __builtin_nontemporal_load(xyz + b3 + 1);
    float pz = __builtin_nontemporal_load(xyz + b3 + 2);

    // pts = xyz * (ws-1), clipped to [0, ws-1]  (XYZ_MIN=0, XYZ_MAX=1)
    px = fminf(fmaxf(px * (float)(WX - 1), 0.0f), (float)(WX - 1));
    py = fminf(fmaxf(py * (float)(WY - 1), 0.0f), (float)(WY - 1));
    pz = fminf(fmaxf(pz * (float)(WZ - 1), 0.0f), (float)(WZ - 1));

    // px >= 0 so truncation == floor
    const int x0 = (int)px, y0 = (int)py, z0 = (int)pz;
    const float fx = px - (float)x0;
    const float fy = py - (float)y0;
    const float fz = pz - (float)z0;
    const int x1 = min(x0 + 1, WX - 1);
    const int y1 = min(y0 + 1, WY - 1);
    const int z1 = min(z0 + 1, WZ - 1);

    const float gx = 1.0f - fx, gy = 1.0f - fy, gz = 1.0f - fz;
    const float w000 = gx * gy * gz, w001 = gx * gy * fz;
    const float w010 = gx * fy * gz, w011 = gx * fy * fz;
    const float w100 = fx * gy * gz, w101 = fx * gy * fz;
    const float w110 = fx * fy * gz, w111 = fx * fy * fz;

    // 8 corner base pointers (offsets fit in 32 bits: grid is 196.6 MB).
    const float* gxp0 = grid + (size_t)(unsigned)(x0 * SX);
    const float* gxp1 = grid + (size_t)(unsigned)(x1 * SX);
    const unsigned oy0 = (unsigned)(y0 * SY), oy1 = (unsigned)(y1 * SY);
    const unsigned oz0 = (unsigned)(z0 * SZ), oz1 = (unsigned)(z1 * SZ);

    const float* c000 = gxp0 + oy0 + oz0;
    const float* c001 = gxp0 + oy0 + oz1;
    const float* c010 = gxp0 + oy1 + oz0;
    const float* c011 = gxp0 + oy1 + oz1;
    const float* c100 = gxp1 + oy0 + oz0;
    const float* c101 = gxp1 + oy0 + oz1;
    const float* c110 = gxp1 + oy1 + oz0;
    const float* c111 = gxp1 + oy1 + oz1;

    // Output row: 12 floats = 48 B, 16 B aligned -> three b128 NT stores.
    float* orow = out + 12ull * (unsigned)i;

    // 12 channels = 3 groups of float4. Corner offsets are multiples of
    // 48 B, so every b128 gather below is 16 B aligned. Fully unrolled:
    // up to 24 global_load_b128 in flight per lane.
#pragma unroll
    for (int g = 0; g < 3; ++g) {
        const v4f v000 = *(const v4f*)(c000 + 4 * g);
        const v4f v001 = *(const v4f*)(c001 + 4 * g);
        const v4f v010 = *(const v4f*)(c010 + 4 * g);
        const v4f v011 = *(const v4f*)(c011 + 4 * g);
        const v4f v100 = *(const v4f*)(c100 + 4 * g);
        const v4f v101 = *(const v4f*)(c101 + 4 * g);
        const v4f v110 = *(const v4f*)(c110 + 4 * g);
        const v4f v111 = *(const v4f*)(c111 + 4 * g);

        v4f acc = v000 * w000;
        acc += v001 * w001;
        acc += v010 * w010;
        acc += v011 * w011;
        acc += v100 * w100;
        acc += v101 * w101;
        acc += v110 * w110;
        acc += v111 * w111;

        // Output is written once and never re-read -> non-temporal store
        // (th:NT) so the 96 MB output stream doesn't evict the grid from L2.
        __builtin_nontemporal_store(acc, (v4f*)(orow + 4 * g));
    }
}

extern "C" void kernel_launch(void* const* d_in, const int* in_sizes, int n_in,
                              void* d_out, int out_size, void* d_ws, size_t ws_size,
                              hipStream_t stream) {
    const float* xyz  = (const float*)d_in[0];   // (N_PTS, 3) f32
    const float* grid = (const float*)d_in[1];   // (160,160,160,12) f32
    float* out = (float*)d_out;                  // (N_PTS, 12) f32

    const int n = in_sizes[0] / 3;               // N_PTS
    const int threads = 256;                     // 8 wave32 per block
    const int blocks = (n + threads - 1) / threads;
    vdb_trilerp12<<<blocks, threads, 0, stream>>>(xyz, grid, out, n);
}